// PairWiseWeightSmoothLoss_84327387890129
// MI455X (gfx1250) — compile-verified
//
#include <hip/hip_runtime.h>
#include <stdint.h>

#define B_DIM 64
#define T_DIM 512
#define V_DIM 512
#define N_ROWS (B_DIM * T_DIM)
#define ROWS_PER_BLK 16
#define PADV (V_DIM + 4)        // 516: stride%64==4 -> conflict-free column reads
#define WAVES 16
#define THREADS (WAVES * 32)
#define RED_BLOCKS 64
#define RED_THREADS 512

typedef float v2f __attribute__((ext_vector_type(2)));
typedef float v8f __attribute__((ext_vector_type(8)));

// Async global -> LDS copy, 16 bytes per lane (GV mode, 64-bit vaddr, ASYNCcnt)
__device__ __forceinline__ void async_load_b128(uint32_t lds_byte_addr, const void* gaddr) {
  asm volatile("global_load_async_to_lds_b128 %0, %1, off"
               :: "v"(lds_byte_addr), "v"(gaddr)
               : "memory");
}
__device__ __forceinline__ void wait_async0() {
  asm volatile("s_wait_asynccnt 0" ::: "memory");
}

extern "C" __global__ __launch_bounds__(THREADS)
void pwws_rows(const float* __restrict__ x,
               const int*   __restrict__ tgt,
               const float* __restrict__ len,
               const float* __restrict__ matric,
               float*       __restrict__ rowloss)
{
  extern __shared__ float smem[];
  float* xb     = smem;                         // 16 x 516  (x rows)
  float* mb     = xb + ROWS_PER_BLK * PADV;     // 16 x 516  (matric rows)
  float* acc    = mb + ROWS_PER_BLK * PADV;     // 256: combined 16x16 D tile
  float* st_lse = acc + 256;                    // 16 each
  float* st_xt  = st_lse + 16;
  float* st_mt  = st_xt  + 16;
  float* st_sM  = st_mt  + 16;
  float* st_sm  = st_sM  + 16;
  int*   st_tg  = (int*)(st_sm + 16);

  const int tid = threadIdx.x;
  const int w   = tid >> 5;     // wave id = row within block
  const int l   = tid & 31;     // lane
  const int n0  = blockIdx.x * ROWS_PER_BLK;
  const int n   = n0 + w;

  if (tid < 256) acc[tid] = 0.0f;

  // ---- async stage: x tile (16 contiguous rows) + per-wave gathered M row ----
  {
    const float* gx = x + (size_t)n0 * V_DIM;
#pragma unroll
    for (int i = 0; i < 4; ++i) {
      int f = i * (THREADS * 4) + tid * 4;      // flat float index in 16x512 tile
      int r = f >> 9, c = f & (V_DIM - 1);
      async_load_b128((uint32_t)(uintptr_t)&xb[r * PADV + c], gx + f);
    }
  }
  const int tg = tgt[n];
  const int fo = ((n & (T_DIM - 1)) == 0) ? 0 : tgt[n - 1];
  {
    const float* gm = matric + ((size_t)fo * V_DIM + tg) * (size_t)V_DIM;
#pragma unroll
    for (int i = 0; i < 4; ++i) {
      int c = i * 128 + l * 4;
      async_load_b128((uint32_t)(uintptr_t)&mb[w * PADV + c], gm + c);
    }
  }
  wait_async0();
  __syncthreads();

  // ---- phase 1: per-row stats (wave w owns row n); lane l covers cols l+32j ----
  float mx = -3.402823466e38f;
#pragma unroll
  for (int j = 0; j < 16; ++j)
    mx = fmaxf(mx, xb[w * PADV + j * 32 + l]);
#pragma unroll
  for (int s = 16; s > 0; s >>= 1) mx = fmaxf(mx, __shfl_xor(mx, s, 32));

  float se = 0.0f, sM = 0.0f;
#pragma unroll
  for (int j = 0; j < 16; ++j) {
    se += expf(xb[w * PADV + j * 32 + l] - mx);
    sM += mb[w * PADV + j * 32 + l];
  }
#pragma unroll
  for (int s = 16; s > 0; s >>= 1) {
    se += __shfl_xor(se, s, 32);
    sM += __shfl_xor(sM, s, 32);
  }
  if (l == 0) {
    float L  = len[n / T_DIM];
    st_lse[w] = mx + logf(se);
    st_xt[w]  = xb[w * PADV + tg];
    st_mt[w]  = mb[w * PADV + tg];
    st_sM[w]  = sM;
    st_sm[w]  = 1.0f - powf(0.9f, 1.0f / L);
    st_tg[w]  = tg;
  }

  // ---- phase 2: D = M(16xK) x X^T(Kx16) via v_wmma_f32_16x16x4_f32,
  //      K=512 partitioned across 16 waves (8 WMMAs each), diag(D)[m]=dot(M_m,x_m)
  {
    const int row  = l & 15;          // A: M-row index; B: X column (= row) index
    const int koff = (l >> 4) * 2;    // lanes 16-31 hold K+2,K+3 per f32 layout
    v8f c;
#pragma unroll
    for (int r = 0; r < 8; ++r) c[r] = 0.0f;
    const int kbase = w * 32;
#pragma unroll
    for (int kk = 0; kk < 32; kk += 4) {
      const int k0 = kbase + kk + koff;
      v2f a, b;
      a[0] = mb[row * PADV + k0];
      a[1] = mb[row * PADV + k0 + 1];
      b[0] = xb[row * PADV + k0];
      b[1] = xb[row * PADV + k0 + 1];
      c = __builtin_amdgcn_wmma_f32_16x16x4_f32(false, a, false, b,
                                                (short)0, c, false, false);
    }
#pragma unroll
    for (int r = 0; r < 8; ++r)
      atomicAdd(&acc[r * 32 + l], c[r]);   // ds_add_f32 combine of partial tiles
  }
  __syncthreads();

  // ---- phase 3: extract diagonal, fold row stats, emit per-row loss ----
  if (tid < ROWS_PER_BLK) {
    const int m   = tid;
    const int idx = (m < 8) ? (m * 33) : ((m - 8) * 32 + m + 16); // D-tile diag
    const float dot = acc[idx];
    const float lse = st_lse[m], xt = st_xt[m], mt = st_mt[m];
    const float SM  = st_sM[m],  sm = st_sm[m];
    const int   t2  = st_tg[m];
    const float s   = sm * SM;
    const float lr  = -(sm * dot - s * lse + (1.0f - s - sm * mt) * (xt - lse));
    rowloss[n0 + m] = (t2 == 0) ? 0.0f : lr;
  }
}

// Stage 1 of deterministic reduction: 64 blocks, each reduces a fixed 512-row
// slice of rowloss and counts PAD targets in that slice.
extern "C" __global__ __launch_bounds__(RED_THREADS)
void pwws_part(const float* __restrict__ rowloss,
               const int*   __restrict__ tgt,
               float*       __restrict__ psum,
               int*         __restrict__ pcnt)
{
  __shared__ float ssum[RED_THREADS];
  __shared__ int   scnt[RED_THREADS];
  const int tid  = threadIdx.x;
  const int base = blockIdx.x * (N_ROWS / RED_BLOCKS);   // 512 rows per block
  const int i    = base + tid;
  ssum[tid] = rowloss[i];
  scnt[tid] = (tgt[i] == 0) ? 1 : 0;
  __syncthreads();
  for (int st = RED_THREADS / 2; st > 0; st >>= 1) {
    if (tid < st) { ssum[tid] += ssum[tid + st]; scnt[tid] += scnt[tid + st]; }
    __syncthreads();
  }
  if (tid == 0) { psum[blockIdx.x] = ssum[0]; pcnt[blockIdx.x] = scnt[0]; }
}

// Stage 2: single tiny block folds the 64 partials, divides, writes the scalar.
extern "C" __global__ __launch_bounds__(RED_BLOCKS)
void pwws_final(const float* __restrict__ psum,
                const int*   __restrict__ pcnt,
                float*       __restrict__ out)
{
  __shared__ float ssum[RED_BLOCKS];
  __shared__ int   scnt[RED_BLOCKS];
  const int tid = threadIdx.x;
  ssum[tid] = psum[tid];
  scnt[tid] = pcnt[tid];
  __syncthreads();
  for (int st = RED_BLOCKS / 2; st > 0; st >>= 1) {
    if (tid < st) { ssum[tid] += ssum[tid + st]; scnt[tid] += scnt[tid + st]; }
    __syncthreads();
  }
  if (tid == 0) out[0] = ssum[0] / (float)scnt[0];
}

extern "C" void kernel_launch(void* const* d_in, const int* in_sizes, int n_in,
                              void* d_out, int out_size, void* d_ws, size_t ws_size,
                              hipStream_t stream) {
  const float* x      = (const float*)d_in[0];
  const int*   tgt    = (const int*)d_in[1];
  const float* len    = (const float*)d_in[2];
  const float* matric = (const float*)d_in[3];

  float* rowloss = (float*)d_ws;               // N_ROWS floats (128 KB)
  float* psum    = rowloss + N_ROWS;           // RED_BLOCKS floats
  int*   pcnt    = (int*)(psum + RED_BLOCKS);  // RED_BLOCKS ints

  const size_t shmem = (size_t)(2 * ROWS_PER_BLK * PADV + 256 + 6 * 16) * sizeof(float);
  hipLaunchKernelGGL(pwws_rows, dim3(N_ROWS / ROWS_PER_BLK), dim3(THREADS), shmem, stream,
                     x, tgt, len, matric, rowloss);
  hipLaunchKernelGGL(pwws_part, dim3(RED_BLOCKS), dim3(RED_THREADS), 0, stream,
                     rowloss, tgt, psum, pcnt);
  hipLaunchKernelGGL(pwws_final, dim3(1), dim3(RED_BLOCKS), 0, stream,
                     psum, pcnt, (float*)d_out);
}